// SparseMHA_81724637708943
// MI455X (gfx1250) — compile-verified
//
#include <hip/hip_runtime.h>
#include <hip/hip_bf16.h>
#include <math.h>

#define HIDDEN    64
#define NUM_HEADS 4
#define HEAD_DIM  16
#define SCALING   0.25f  // HEAD_DIM^-0.5

typedef __attribute__((ext_vector_type(2))) float v2f;
typedef __attribute__((ext_vector_type(8))) float v8f;

// ---------------------------------------------------------------------------
// Kernel 0: zero the output (harness poisons d_out; scatter needs zeros)
// ---------------------------------------------------------------------------
__global__ void SparseMHA_zero_kernel(float4* __restrict__ out, int n4) {
    int i = blockIdx.x * blockDim.x + threadIdx.x;
    if (i < n4) out[i] = make_float4(0.f, 0.f, 0.f, 0.f);
}

// ---------------------------------------------------------------------------
// Kernel 1: fused q/k/v projection using native fp32 WMMA (16x16x4).
//   Block = 16 rows of h, 128 threads = 4 waves, wave w -> columns [16w,16w+16).
//   Each wave runs 16 K-steps; A fragment shared across the 3 weight matrices,
//   so 48 v_wmma_f32_16x16x4_f32 per wave. Epilogue adds bias (and applies
//   SCALING to q as in the reference).
// ---------------------------------------------------------------------------
__global__ __launch_bounds__(128) void SparseMHA_proj_kernel(
    const float* __restrict__ h,
    const float* __restrict__ Wq, const float* __restrict__ bq,
    const float* __restrict__ Wk, const float* __restrict__ bk,
    const float* __restrict__ Wv, const float* __restrict__ bv,
    float* __restrict__ qout, float* __restrict__ kout, float* __restrict__ vout,
    int n)
{
    const int m0   = blockIdx.x * 16;
    const int wave = threadIdx.x >> 5;
    const int lane = threadIdx.x & 31;
    const int half = lane >> 4;      // 0: K lo pair / M,N 0-7 side, 1: K hi pair
    const int r16  = lane & 15;
    const int n0   = wave * 16;

    // Clamp A-row so WMMA can run with EXEC all-ones even on a ragged tail.
    const int arow = (m0 + r16 < n) ? (m0 + r16) : (n - 1);

    const float* __restrict__ hrow  = h  + arow * HIDDEN;
    const float* __restrict__ wqrow = Wq + (n0 + r16) * HIDDEN;  // B[kk,n] = W[n0+n, kb+kk]
    const float* __restrict__ wkrow = Wk + (n0 + r16) * HIDDEN;
    const float* __restrict__ wvrow = Wv + (n0 + r16) * HIDDEN;

    v8f accq = {}; v8f acck = {}; v8f accv = {};

#pragma unroll
    for (int ks = 0; ks < 16; ++ks) {
        const int kb = ks * 4 + half * 2;   // this lane's K pair within the 4-wide slab
        v2f afrag = *(const v2f*)(hrow  + kb);
        v2f bqf   = *(const v2f*)(wqrow + kb);
        v2f bkf   = *(const v2f*)(wkrow + kb);
        v2f bvf   = *(const v2f*)(wvrow + kb);
        accq = __builtin_amdgcn_wmma_f32_16x16x4_f32(false, afrag, false, bqf,
                                                     (short)0, accq, false, false);
        acck = __builtin_amdgcn_wmma_f32_16x16x4_f32(false, afrag, false, bkf,
                                                     (short)0, acck, false, false);
        accv = __builtin_amdgcn_wmma_f32_16x16x4_f32(false, afrag, false, bvf,
                                                     (short)0, accv, false, false);
    }

    // C layout: VGPR vr, lanes 0-15 -> M = vr, lanes 16-31 -> M = vr + 8; N = lane%16
    const int   colj  = n0 + r16;
    const float biasq = bq[colj];
    const float biask = bk[colj];
    const float biasv = bv[colj];

#pragma unroll
    for (int vr = 0; vr < 8; ++vr) {
        const int m = m0 + vr + half * 8;
        if (m < n) {
            const int idx = m * HIDDEN + colj;
            qout[idx] = (accq[vr] + biasq) * SCALING;
            kout[idx] =  acck[vr] + biask;
            vout[idx] =  accv[vr] + biasv;
        }
    }
}

// ---------------------------------------------------------------------------
// Kernel 2: fused SDDMM + sigmoid*edge_weight + SPMM scatter.
//   16 lanes per edge: lane l covers channels [4l, 4l+4) = (d=l, heads 0..3).
//   256 B contiguous gathers of q[row]/k[col]/v[col]; all-lane butterfly
//   reduction over d (width-16 shfl_xor on wave32); float atomics to out[row].
//   Working set (q,k,v,out = 51 MB) is fully resident in the 192 MB L2.
// ---------------------------------------------------------------------------
__global__ __launch_bounds__(256) void SparseMHA_edge_kernel(
    const float* __restrict__ q, const float* __restrict__ k,
    const float* __restrict__ v,
    const int* __restrict__ row, const int* __restrict__ col,
    const float* __restrict__ ew,
    float* __restrict__ out, int nnz)
{
    const int lane = threadIdx.x & 31;
    const int wave = threadIdx.x >> 5;
    const int sub  = lane >> 4;      // which of the wave's 2 edges
    const int l    = lane & 15;      // d index / float4 slot within the node vector

    const int e = blockIdx.x * 16 + wave * 2 + sub;
    if (e >= nnz) return;            // whole 16-lane group exits together

    const int r = row[e];
    const int c = col[e];

    const float4 q4 = *(const float4*)(q + r * HIDDEN + l * 4);
    const float4 k4 = *(const float4*)(k + c * HIDDEN + l * 4);

    float p0 = q4.x * k4.x;
    float p1 = q4.y * k4.y;
    float p2 = q4.z * k4.z;
    float p3 = q4.w * k4.w;

#pragma unroll
    for (int off = 8; off >= 1; off >>= 1) {   // butterfly: every lane gets full sums
        p0 += __shfl_xor(p0, off, 16);
        p1 += __shfl_xor(p1, off, 16);
        p2 += __shfl_xor(p2, off, 16);
        p3 += __shfl_xor(p3, off, 16);
    }

    const float w  = ew[e];
    const float a0 = w / (1.0f + __expf(-p0));
    const float a1 = w / (1.0f + __expf(-p1));
    const float a2 = w / (1.0f + __expf(-p2));
    const float a3 = w / (1.0f + __expf(-p3));

    const float4 v4 = *(const float4*)(v + c * HIDDEN + l * 4);
    float* __restrict__ ob = out + r * HIDDEN + l * 4;
    atomicAdd(ob + 0, a0 * v4.x);
    atomicAdd(ob + 1, a1 * v4.y);
    atomicAdd(ob + 2, a2 * v4.z);
    atomicAdd(ob + 3, a3 * v4.w);
}

// ---------------------------------------------------------------------------
extern "C" void kernel_launch(void* const* d_in, const int* in_sizes, int n_in,
                              void* d_out, int out_size, void* d_ws, size_t ws_size,
                              hipStream_t stream) {
    const float* h   = (const float*)d_in[0];
    const int*   row = (const int*)  d_in[1];
    const int*   col = (const int*)  d_in[2];
    const float* ew  = (const float*)d_in[3];
    const float* Wq  = (const float*)d_in[4];
    const float* bq  = (const float*)d_in[5];
    const float* Wk  = (const float*)d_in[6];
    const float* bk  = (const float*)d_in[7];
    const float* Wv  = (const float*)d_in[8];
    const float* bv  = (const float*)d_in[9];
    float*       out = (float*)d_out;

    const int n   = in_sizes[0] / HIDDEN;   // 50000
    const int nnz = in_sizes[1];            // 1600000

    float* qws = (float*)d_ws;
    float* kws = qws + (size_t)n * HIDDEN;
    float* vws = kws + (size_t)n * HIDDEN;

    // zero d_out (poisoned by harness)
    const int n4 = (n * HIDDEN) / 4;
    SparseMHA_zero_kernel<<<(n4 + 255) / 256, 256, 0, stream>>>((float4*)out, n4);

    // q/k/v projections via fp32 WMMA
    SparseMHA_proj_kernel<<<(n + 15) / 16, 128, 0, stream>>>(
        h, Wq, bq, Wk, bk, Wv, bv, qws, kws, vws, n);

    // fused edge SDDMM + sigmoid*w + scatter (16 edges per 256-thread block)
    SparseMHA_edge_kernel<<<(nnz + 15) / 16, 256, 0, stream>>>(
        qws, kws, vws, row, col, ew, out, nnz);
}